// SupervisedBCEWithGraphConsistency_62543313764409
// MI455X (gfx1250) — compile-verified
//
#include <hip/hip_runtime.h>

// SupervisedBCEWithGraphConsistency — MI455X / gfx1250
//
// Strategy: factor the (4096 x 64 x 128) neighbor reduction into per-block
// precomputed sums (s_blk = sum p*keep, c_blk = sum keep), reducing gather
// work 128x. The remaining 16-blocks x K=64 masked gather row-sum is done on
// the matrix pipe with V_WMMA_F32_16X16X4_F32 (B = ones => D row sums), which
// is exact FP32 FMA (RNE) and matches the reference numerics.
// Branchless masking in the WMMA loop (indices are always in-range) keeps
// EXEC all-1s and lets the gathers pipeline.
// No atomics: per-block partials + fixed-shape tree reduce => deterministic.

typedef __attribute__((ext_vector_type(2))) float v2f;
typedef __attribute__((ext_vector_type(8))) float v8f;

#define BLOCK 128

__device__ __forceinline__ float softplus_f(float x) {
    // stable: max(x,0) + log1p(exp(-|x|))
    return fmaxf(x, 0.0f) + log1pf(expf(-fabsf(x)));
}
__device__ __forceinline__ float sigmoid_f(float x) {
    return 1.0f / (1.0f + expf(-x));
}
__device__ __forceinline__ float wave_reduce_sum(float v) {
    #pragma unroll
    for (int off = 16; off > 0; off >>= 1)
        v += __shfl_down(v, off, 32);
    return v;
}

// ---------------------------------------------------------------------------
// Pass A: elementwise BCE/sigmoid + per-block sums. One 128-thread WG / block.
// ---------------------------------------------------------------------------
__global__ void pass_a_kernel(const float* __restrict__ logits,
                              const float* __restrict__ targets,
                              const unsigned char* __restrict__ sup_mask,
                              const unsigned char* __restrict__ ign_mask,
                              float* __restrict__ p_out,
                              float* __restrict__ s_blk,
                              float* __restrict__ c_blk,
                              float* __restrict__ q_cnt,
                              float* __restrict__ bce_part,
                              float* __restrict__ sup_part) {
    const int b = blockIdx.x;
    const int t = threadIdx.x;
    const int i = b * BLOCK + t;

    const float x  = logits[i];
    const float tg = targets[i];
    const bool sup = sup_mask[i] != 0;
    const bool ign = ign_mask[i] != 0;

    const float p = sigmoid_f(x);
    p_out[i] = p;

    const float bce    = softplus_f(x) - tg * x;
    const float sup_f  = sup ? 1.0f : 0.0f;
    const float keep_f = ign ? 0.0f : 1.0f;
    const float unc_f  = (!ign && !sup) ? 1.0f : 0.0f;

    float v0 = bce * sup_f;   // sup-masked bce sum
    float v1 = sup_f;         // sup count
    float v2 = p * keep_f;    // s_blk
    float v3 = keep_f;        // c_blk
    float v4 = unc_f;         // q_cnt

    v0 = wave_reduce_sum(v0);
    v1 = wave_reduce_sum(v1);
    v2 = wave_reduce_sum(v2);
    v3 = wave_reduce_sum(v3);
    v4 = wave_reduce_sum(v4);

    __shared__ float lds[4][5];
    const int lane = t & 31, wid = t >> 5;
    if (lane == 0) {
        lds[wid][0] = v0; lds[wid][1] = v1; lds[wid][2] = v2;
        lds[wid][3] = v3; lds[wid][4] = v4;
    }
    __syncthreads();
    if (t == 0) {
        float s0 = 0.f, s1 = 0.f, s2 = 0.f, s3 = 0.f, s4 = 0.f;
        #pragma unroll
        for (int w = 0; w < 4; ++w) {
            s0 += lds[w][0]; s1 += lds[w][1]; s2 += lds[w][2];
            s3 += lds[w][3]; s4 += lds[w][4];
        }
        bce_part[b] = s0;
        sup_part[b] = s1;
        s_blk[b]    = s2;
        c_blk[b]    = s3;
        q_cnt[b]    = s4;
    }
}

// ---------------------------------------------------------------------------
// Pass B: 16 blocks per WG (256 threads = 8 waves).
//   Wave 0: WMMA f32 16x16x4 masked gather row-sum over K = maxkv neighbors,
//           yielding n_sum / n_cnt for the 16 blocks (M dimension).
//   All waves: per-block squared-deviation sum over 128 elements.
// ---------------------------------------------------------------------------
__global__ void pass_b_kernel(const float* __restrict__ p,
                              const unsigned char* __restrict__ sup_mask,
                              const unsigned char* __restrict__ ign_mask,
                              const int* __restrict__ kv_indices,
                              const int* __restrict__ kv_num,
                              const float* __restrict__ s_blk,
                              const float* __restrict__ c_blk,
                              const float* __restrict__ q_cnt,
                              float* __restrict__ graph_part,
                              float* __restrict__ valid_part,
                              int maxkv) {
    const int b0   = blockIdx.x * 16;
    const int t    = threadIdx.x;
    const int lane = t & 31;
    const int wid  = t >> 5;

    __shared__ float ls_nmean[16];
    __shared__ float ls_ncnt[16];

    if (wid == 0) {
        // A-matrix 16x4 f32 layout: lanes 0-15 hold {K=0,K=1}; lanes 16-31 {K=2,K=3}.
        const int m     = lane & 15;
        const int khalf = (lane >> 4) * 2;          // 0 or 2
        const int b     = b0 + m;
        const int kvn   = kv_num[b];
        const int base  = b * maxkv;

        v8f acc_s = {};
        v8f acc_c = {};
        const v2f ones = {1.0f, 1.0f};              // B = all-ones => row sums

        for (int kk = 0; kk < maxkv; kk += 4) {
            const int k0 = kk + khalf;              // even => int2 is 8B-aligned
            const int2 idx2 = *(const int2*)(kv_indices + base + k0);
            // Branchless validity mask: gathers are always in-range, only the
            // contribution is masked (kv_valid: k < kv_num_blocks[b]).
            const float m0 = (k0     < kvn) ? 1.0f : 0.0f;
            const float m1 = (k0 + 1 < kvn) ? 1.0f : 0.0f;
            v2f a_s, a_c;
            a_s.x = s_blk[idx2.x] * m0;
            a_s.y = s_blk[idx2.y] * m1;
            a_c.x = c_blk[idx2.x] * m0;
            a_c.y = c_blk[idx2.y] * m1;
            // D = A x ones + C : every column of D holds the K-partial row sums
            acc_s = __builtin_amdgcn_wmma_f32_16x16x4_f32(
                false, a_s, false, ones, (short)0, acc_s, false, false);
            acc_c = __builtin_amdgcn_wmma_f32_16x16x4_f32(
                false, a_c, false, ones, (short)0, acc_c, false, false);
        }
        // C/D layout: lanes 0-15 vgpr r -> M=r (N=lane); lanes 16-31 -> M=8+r.
        if (lane == 0) {
            #pragma unroll
            for (int r = 0; r < 8; ++r) {
                const float nc = acc_c[r];
                ls_nmean[r] = acc_s[r] / fmaxf(nc, 1.0f);
                ls_ncnt[r]  = nc;
            }
        }
        if (lane == 16) {
            #pragma unroll
            for (int r = 0; r < 8; ++r) {
                const float nc = acc_c[r];
                ls_nmean[8 + r] = acc_s[r] / fmaxf(nc, 1.0f);
                ls_ncnt[8 + r]  = nc;
            }
        }
    }
    __syncthreads();

    // Each wave handles 2 of the 16 blocks: 128 elements / 32 lanes.
    #pragma unroll
    for (int s = 0; s < 2; ++s) {
        const int m  = wid * 2 + s;
        const int b  = b0 + m;
        const float nm = ls_nmean[m];
        float sum = 0.0f;
        #pragma unroll
        for (int eo = 0; eo < BLOCK; eo += 32) {
            const int i = b * BLOCK + eo + lane;
            const bool sup = sup_mask[i] != 0;
            const bool ign = ign_mask[i] != 0;
            if (!ign && !sup) {
                const float d = p[i] - nm;
                sum += d * d;
            }
        }
        sum = wave_reduce_sum(sum);
        if (lane == 0) {
            const float qc       = q_cnt[b];
            const float blk_loss = sum / fmaxf(qc, 1.0f);
            const float valid    = (qc > 0.0f && ls_ncnt[m] > 0.0f) ? 1.0f : 0.0f;
            graph_part[b] = blk_loss * valid;
            valid_part[b] = valid;
        }
    }
}

// ---------------------------------------------------------------------------
// Finalize: single WG deterministic tree reduction of the 4 partial arrays.
// ---------------------------------------------------------------------------
__global__ void finalize_kernel(const float* __restrict__ bce_part,
                                const float* __restrict__ sup_part,
                                const float* __restrict__ graph_part,
                                const float* __restrict__ valid_part,
                                float* __restrict__ out, int nb) {
    __shared__ float lds[256][4];
    const int t = threadIdx.x;
    float s0 = 0.f, s1 = 0.f, s2 = 0.f, s3 = 0.f;
    for (int i = t; i < nb; i += 256) {
        s0 += bce_part[i];
        s1 += sup_part[i];
        s2 += graph_part[i];
        s3 += valid_part[i];
    }
    lds[t][0] = s0; lds[t][1] = s1; lds[t][2] = s2; lds[t][3] = s3;
    __syncthreads();
    for (int off = 128; off > 0; off >>= 1) {
        if (t < off) {
            lds[t][0] += lds[t + off][0];
            lds[t][1] += lds[t + off][1];
            lds[t][2] += lds[t + off][2];
            lds[t][3] += lds[t + off][3];
        }
        __syncthreads();
    }
    if (t == 0) {
        const float S_bce = lds[0][0];
        const float S_sup = lds[0][1];
        const float S_g   = lds[0][2];
        const float S_v   = lds[0][3];
        const float loss_sup   = (S_sup > 0.0f) ? (S_bce / fmaxf(S_sup, 1.0f)) : 0.0f;
        const float loss_graph = S_g / fmaxf(S_v, 1.0f);
        out[0] = loss_sup + 0.3f * loss_graph;
    }
}

extern "C" void kernel_launch(void* const* d_in, const int* in_sizes, int n_in,
                              void* d_out, int out_size, void* d_ws, size_t ws_size,
                              hipStream_t stream) {
    const float*         logits  = (const float*)d_in[0];
    const float*         targets = (const float*)d_in[1];
    const unsigned char* supm    = (const unsigned char*)d_in[2];  // jnp bool = u8
    const unsigned char* ignm    = (const unsigned char*)d_in[3];  // jnp bool = u8
    const int*           kv_idx  = (const int*)d_in[4];
    const int*           kv_num  = (const int*)d_in[5];
    // d_in[6] = block_size scalar (128); BLOCK is compile-time here.

    const int n     = in_sizes[0];           // 524288
    const int nb    = n / BLOCK;             // 4096
    const int maxkv = in_sizes[4] / nb;      // 64

    float* wsf        = (float*)d_ws;
    float* p          = wsf;                 // n
    float* s_blk      = p + n;               // nb
    float* c_blk      = s_blk + nb;          // nb
    float* q_cnt      = c_blk + nb;          // nb
    float* bce_part   = q_cnt + nb;          // nb
    float* sup_part   = bce_part + nb;       // nb
    float* graph_part = sup_part + nb;       // nb
    float* valid_part = graph_part + nb;     // nb

    pass_a_kernel<<<nb, BLOCK, 0, stream>>>(logits, targets, supm, ignm,
                                            p, s_blk, c_blk, q_cnt,
                                            bce_part, sup_part);
    pass_b_kernel<<<nb / 16, 256, 0, stream>>>(p, supm, ignm, kv_idx, kv_num,
                                               s_blk, c_blk, q_cnt,
                                               graph_part, valid_part, maxkv);
    finalize_kernel<<<1, 256, 0, stream>>>(bce_part, sup_part,
                                           graph_part, valid_part,
                                           (float*)d_out, nb);
}